// Layer_44126493999491
// MI455X (gfx1250) — compile-verified
//
#include <hip/hip_runtime.h>
#include <hip/hip_bf16.h>

#define N_NODES 10000
#define N_EDGES 160000
#define MUL 64

// ext-vector types for WMMA operands
typedef __attribute__((ext_vector_type(2))) float v2f;
typedef __attribute__((ext_vector_type(8))) float v8f;

// ---- constants from the reference ----
#define SQRT3_F      1.7320508075688772f
#define INV_SQRT3_F  0.5773502691896258f
#define INV_SQRT2_F  0.7071067811865476f
// scales applied to A fragments (workspace holds RAW segment sums, so the
// reference's /DENOM(=16) is folded in):
#define SLIN0 (1.0f / (16.0f * 13.856406460551018f))  // 1/(16*sqrt(192))
#define SLIN1 (1.0f / 256.0f)                          // 1/(16*sqrt(256))
#define SSC   (0.125f)                                 // 1/sqrt(64)

// Native non-returning f32 atomic add, device scope (cross-WGP reduction).
// Guarantees GLOBAL_ATOMIC_ADD_F32 (STOREcnt-tracked) instead of a CAS loop.
__device__ __forceinline__ void fadd(float* p, float v) {
  asm volatile("global_atomic_add_f32 %0, %1, off scope:SCOPE_DEV"
               :: "v"(p), "v"(v)
               : "memory");
}

// Async global->LDS copy of 16 bytes per lane (ASYNCcnt-tracked, no VGPR data).
// lds_off = LDS byte address (low 32 bits of the generic shared pointer).
__device__ __forceinline__ void async_copy_b128(unsigned lds_off, const void* gptr) {
  asm volatile("global_load_async_to_lds_b128 %0, %1, off"
               :: "v"(lds_off), "v"(gptr)
               : "memory");
}

__device__ __forceinline__ void wait_async_all() {
  asm volatile("s_wait_asynccnt 0x0" ::: "memory");
}

// ------------------------------------------------------------------
// Kernel 0: zero the accumulator workspace (9.6M floats)
// ------------------------------------------------------------------
__global__ __launch_bounds__(256) void zero_ws_kernel(float* __restrict__ ws, int n) {
  int i = blockIdx.x * blockDim.x + threadIdx.x;
  int stride = gridDim.x * blockDim.x;
  for (; i < n; i += stride) ws[i] = 0.0f;
}

// ------------------------------------------------------------------
// Kernel 1: edge phase. One wave32 per edge; lane l handles channels
// c = l and c = l + 32. Raw segment sums via native f32 atomics into
//   a0: [node][192]
//   a1: [comp i][node][256]   (component-major so GEMM A-loads are unit-stride)
// m0 = [s0 | s0 | (s1.y1)/sqrt3]
// m1 blocks (per component): [s1 | s0*y1 | s1 | cross(s1,y1)/sqrt2]
// ------------------------------------------------------------------
__global__ __launch_bounds__(256) void edge_scatter_kernel(
    const float* __restrict__ pos, const float* __restrict__ nf,
    const int* __restrict__ snd, const int* __restrict__ rcv,
    float* __restrict__ a0, float* __restrict__ a1) {
  const int edge = (blockIdx.x * blockDim.x + threadIdx.x) >> 5;
  const int lane = threadIdx.x & 31;
  if (edge >= N_EDGES) return;

  const int s = snd[edge];
  const int r = rcv[edge];

  // edge direction (all lanes redundantly; broadcast L2 hits)
  const float rx = pos[r * 3 + 0] - pos[s * 3 + 0];
  const float ry = pos[r * 3 + 1] - pos[s * 3 + 1];
  const float rz = pos[r * 3 + 2] - pos[s * 3 + 2];
  const float inv = rsqrtf(rx * rx + ry * ry + rz * rz);
  float y1[3];
  y1[0] = SQRT3_F * rx * inv;
  y1[1] = SQRT3_F * ry * inv;
  y1[2] = SQRT3_F * rz * inv;

  const float* __restrict__ srow = nf + (size_t)s * (4 * MUL);
  float* __restrict__ A0 = a0 + (size_t)r * 192;

#pragma unroll
  for (int t = 0; t < 2; ++t) {
    const int c = lane + 32 * t;  // channel 0..63, coalesced s0 loads
    const float s0 = srow[c];
    float s1[3];
    s1[0] = srow[MUL + 3 * c + 0];
    s1[1] = srow[MUL + 3 * c + 1];
    s1[2] = srow[MUL + 3 * c + 2];

    const float dot = (s1[0] * y1[0] + s1[1] * y1[1] + s1[2] * y1[2]) * INV_SQRT3_F;
    float cr[3];  // cross(s1, y1) * 1/sqrt2
    cr[0] = (s1[1] * y1[2] - s1[2] * y1[1]) * INV_SQRT2_F;
    cr[1] = (s1[2] * y1[0] - s1[0] * y1[2]) * INV_SQRT2_F;
    cr[2] = (s1[0] * y1[1] - s1[1] * y1[0]) * INV_SQRT2_F;

    // a0: m0 = [s0 | s0 | dot]
    fadd(A0 + c, s0);
    fadd(A0 + 64 + c, s0);
    fadd(A0 + 128 + c, dot);

    // a1 (component-major layout)
#pragma unroll
    for (int i = 0; i < 3; ++i) {
      float* __restrict__ A1 =
          a1 + (size_t)i * ((size_t)N_NODES * 256) + (size_t)r * 256;
      fadd(A1 + c, s1[i]);            // block 0: s1
      fadd(A1 + 64 + c, s0 * y1[i]);  // block 1: s0 (x) y1
      fadd(A1 + 128 + c, s1[i]);      // block 2: s1
      fadd(A1 + 192 + c, cr[i]);      // block 3: cross/sqrt2
    }
  }
}

// ------------------------------------------------------------------
// WMMA f32 16x16x4 fragment mapping (wave32):
//   A: lane L -> row = L&15, K-pair offset koff = (L>=16)?2:0
//   B: lane L -> col = L&15, same koff
//   C/D: VGPR v -> row = v + 8*(L>=16), col = L&15
// Weights are staged RAW into LDS with async B128 copies; the segment
// scales are applied to the A fragments (VALU co-executes with WMMA).
// ------------------------------------------------------------------

// stage `nfloat4` float4s from g into LDS starting at lds byte offset base.
__device__ __forceinline__ void stage_weights_async(float* sW, unsigned lds_base_f,
                                                    const float* __restrict__ g,
                                                    int nfloat4) {
  const unsigned lds_base = (unsigned)(uintptr_t)(sW) + lds_base_f * 4u;
  for (int idx = threadIdx.x; idx < nfloat4; idx += 256) {
    async_copy_b128(lds_base + (unsigned)idx * 16u, g + (size_t)idx * 4);
  }
}

// Kernel 2: out0 = [a0*SLIN0 | x0*SSC] (K=256) @ [W_lin0 ; W_sc0] -> out[:,0:64]
__global__ __launch_bounds__(256) void gemm_out0_kernel(
    const float* __restrict__ a0, const float* __restrict__ nf,
    const float* __restrict__ Wlin0, const float* __restrict__ Wsc0,
    float* __restrict__ out) {
  extern __shared__ float sW[];  // 256*64 floats = 64 KB, raw weights
  stage_weights_async(sW, 0, Wlin0, (192 * 64) / 4);
  stage_weights_async(sW, 192 * 64, Wsc0, (64 * 64) / 4);
  wait_async_all();
  __syncthreads();

  const int lane = threadIdx.x & 31;
  const int wave = threadIdx.x >> 5;
  const int cw = wave & 3;            // col tile 0..3
  const int rw = wave >> 2;           // row sub-tile 0..1
  const int mrow = lane & 15;
  const int koff = (lane >> 4) << 1;  // 0 or 2
  const int colg = cw * 16 + (lane & 15);

  for (int tile = blockIdx.x * 2 + rw; tile < 625; tile += gridDim.x * 2) {
    const float* __restrict__ Aa = a0 + (size_t)(tile * 16 + mrow) * 192;
    const float* __restrict__ Ax = nf + (size_t)(tile * 16 + mrow) * 256;  // x0
    v8f acc = {};
#pragma unroll 8
    for (int kb = 0; kb < 192; kb += 4) {
      v2f a, b;
      a.x = Aa[kb + koff] * SLIN0;
      a.y = Aa[kb + koff + 1] * SLIN0;
      b.x = sW[(kb + koff) * 64 + colg];
      b.y = sW[(kb + koff + 1) * 64 + colg];
      acc = __builtin_amdgcn_wmma_f32_16x16x4_f32(false, a, false, b, (short)0, acc,
                                                  false, false);
    }
#pragma unroll 8
    for (int kb = 0; kb < 64; kb += 4) {
      v2f a, b;
      a.x = Ax[kb + koff] * SSC;
      a.y = Ax[kb + koff + 1] * SSC;
      b.x = sW[(192 + kb + koff) * 64 + colg];
      b.y = sW[(192 + kb + koff + 1) * 64 + colg];
      acc = __builtin_amdgcn_wmma_f32_16x16x4_f32(false, a, false, b, (short)0, acc,
                                                  false, false);
    }
    const int rbase = tile * 16 + ((lane >> 4) << 3);
#pragma unroll
    for (int v = 0; v < 8; ++v)
      out[(size_t)(rbase + v) * 256 + colg] = acc[v];
  }
}

// Kernel 3: per component i (blockIdx.y):
//   out1_i = [a1_i*SLIN1 | x1_i*SSC] (K=320) @ [W_lin1 ; W_sc1]
//   written to out[:, 64 + 3*k + i]
__global__ __launch_bounds__(256) void gemm_out1_kernel(
    const float* __restrict__ a1, const float* __restrict__ nf,
    const float* __restrict__ Wlin1, const float* __restrict__ Wsc1,
    float* __restrict__ out) {
  extern __shared__ float sW[];  // 320*64 floats = 80 KB (<< 320 KB/WGP)
  stage_weights_async(sW, 0, Wlin1, (256 * 64) / 4);
  stage_weights_async(sW, 256 * 64, Wsc1, (64 * 64) / 4);
  wait_async_all();
  __syncthreads();

  const int comp = blockIdx.y;  // 0..2
  const int lane = threadIdx.x & 31;
  const int wave = threadIdx.x >> 5;
  const int cw = wave & 3;
  const int rw = wave >> 2;
  const int mrow = lane & 15;
  const int koff = (lane >> 4) << 1;
  const int colg = cw * 16 + (lane & 15);

  const float* __restrict__ a1c = a1 + (size_t)comp * ((size_t)N_NODES * 256);

  for (int tile = blockIdx.x * 2 + rw; tile < 625; tile += gridDim.x * 2) {
    const float* __restrict__ Aa = a1c + (size_t)(tile * 16 + mrow) * 256;  // unit stride
    const float* __restrict__ Ax =
        nf + (size_t)(tile * 16 + mrow) * 256 + 64 + comp;  // stride 3
    v8f acc = {};
#pragma unroll 8
    for (int kb = 0; kb < 256; kb += 4) {
      v2f a, b;
      a.x = Aa[kb + koff] * SLIN1;
      a.y = Aa[kb + koff + 1] * SLIN1;
      b.x = sW[(kb + koff) * 64 + colg];
      b.y = sW[(kb + koff + 1) * 64 + colg];
      acc = __builtin_amdgcn_wmma_f32_16x16x4_f32(false, a, false, b, (short)0, acc,
                                                  false, false);
    }
#pragma unroll 8
    for (int kb = 0; kb < 64; kb += 4) {
      v2f a, b;
      a.x = Ax[3 * (kb + koff)] * SSC;
      a.y = Ax[3 * (kb + koff) + 3] * SSC;
      b.x = sW[(256 + kb + koff) * 64 + colg];
      b.y = sW[(256 + kb + koff + 1) * 64 + colg];
      acc = __builtin_amdgcn_wmma_f32_16x16x4_f32(false, a, false, b, (short)0, acc,
                                                  false, false);
    }
    const int rbase = tile * 16 + ((lane >> 4) << 3);
#pragma unroll
    for (int v = 0; v < 8; ++v)
      out[(size_t)(rbase + v) * 256 + 64 + 3 * colg + comp] = acc[v];
  }
}

// ------------------------------------------------------------------
extern "C" void kernel_launch(void* const* d_in, const int* in_sizes, int n_in,
                              void* d_out, int out_size, void* d_ws, size_t ws_size,
                              hipStream_t stream) {
  const float* pos   = (const float*)d_in[0];
  const float* nf    = (const float*)d_in[1];
  const int*   snd   = (const int*)d_in[2];
  const int*   rcv   = (const int*)d_in[3];
  const float* Wlin0 = (const float*)d_in[4];
  const float* Wlin1 = (const float*)d_in[5];
  const float* Wsc0  = (const float*)d_in[6];
  const float* Wsc1  = (const float*)d_in[7];
  float* out = (float*)d_out;

  float* ws = (float*)d_ws;
  float* a0 = ws;                              // [N][192]
  float* a1 = ws + (size_t)N_NODES * 192;      // [3][N][256]
  const int ws_elems = N_NODES * (192 + 768);  // 9.6M floats

  zero_ws_kernel<<<2048, 256, 0, stream>>>(ws, ws_elems);

  // one wave32 per edge -> 160000 waves -> 20000 blocks of 256
  edge_scatter_kernel<<<(N_EDGES * 32) / 256, 256, 0, stream>>>(pos, nf, snd, rcv,
                                                                a0, a1);

  gemm_out0_kernel<<<120, 256, 256 * 64 * sizeof(float), stream>>>(a0, nf, Wlin0,
                                                                   Wsc0, out);
  gemm_out1_kernel<<<dim3(120, 3), 256, 320 * 64 * sizeof(float), stream>>>(
      a1, nf, Wlin1, Wsc1, out);
}